// GNN_11527692223233
// MI455X (gfx1250) — compile-verified
//
#include <hip/hip_runtime.h>
#include <cstddef>

typedef __attribute__((ext_vector_type(2))) float v2f;
typedef __attribute__((ext_vector_type(8))) float v8f;

#define D_IN 128

// ---------------- degree / normalization ----------------

__global__ void deg_init_kernel(float* deg, int n) {
    int i = blockIdx.x * blockDim.x + threadIdx.x;
    if (i < n) deg[i] = 1.0f;               // self-loop contributes 1
}

__global__ void deg_scatter_kernel(const int* __restrict__ dst, float* deg, int E) {
    int e = blockIdx.x * blockDim.x + threadIdx.x;
    if (e < E) atomicAdd(&deg[dst[e]], 1.0f);
}

__global__ void rsqrt_kernel(float* deg, int n) {
    int i = blockIdx.x * blockDim.x + threadIdx.x;
    if (i < n) deg[i] = rsqrtf(deg[i]);     // deg >= 1 always
}

// ---------------- fp32 WMMA GEMM: out[N,FOUT] = H[N,FIN] @ W[FIN,FOUT] ----------------
// One wave computes one 16x16 tile via V_WMMA_F32_16X16X4_F32, K stepped by 4.
// A-layout (16x4 f32): lanes 0-15 -> M=lane, K={k,k+1}; lanes 16-31 -> M=lane-16, K={k+2,k+3}
// B-layout (4x16 f32): VGPR0 -> rows k (lanes 0-15) / k+2 (lanes 16-31); VGPR1 -> k+1 / k+3
// C/D (16x16 f32): VGPR i -> M=i (lanes 0-15), M=i+8 (lanes 16-31), N=lane&15

template<int FIN, int FOUT>
__global__ __launch_bounds__(256)
void gcn_gemm_wmma(const float* __restrict__ H, const float* __restrict__ W,
                   float* __restrict__ out, int n) {
    constexpr int COLT   = FOUT / 16;       // column tiles per block
    constexpr int ROWT   = 8 / COLT;        // row tiles per block (8 waves total)
    constexpr int ROWS   = ROWT * 16;
    constexpr int STRIDE = FIN + 4;         // LDS pad: break bank conflicts on column reads
    __shared__ float lds[ROWS * STRIDE];

    const int row0 = blockIdx.x * ROWS;

    // stage A tile (coalesced), zero-fill rows past N
    for (int i = threadIdx.x; i < ROWS * FIN; i += 256) {
        int r = i / FIN, cidx = i - r * FIN;
        int g = row0 + r;
        lds[r * STRIDE + cidx] = (g < n) ? H[(size_t)g * FIN + cidx] : 0.0f;
    }
    __syncthreads();

    const int w    = threadIdx.x >> 5;
    const int lane = threadIdx.x & 31;
    const int rt   = w / COLT;
    const int ct   = w - rt * COLT;
    const int half = lane >> 4;             // 0: lanes 0-15, 1: lanes 16-31
    const int l15  = lane & 15;
    const int kb   = half * 2;

    const float* lrow = &lds[(rt * 16 + l15) * STRIDE];
    const float* wcol = &W[ct * 16 + l15];

    v8f c = {0.f, 0.f, 0.f, 0.f, 0.f, 0.f, 0.f, 0.f};
    #pragma unroll
    for (int k = 0; k < FIN; k += 4) {
        v2f a, b;
        a.x = lrow[k + kb];
        a.y = lrow[k + kb + 1];
        b.x = wcol[(size_t)(k + kb) * FOUT];
        b.y = wcol[(size_t)(k + kb + 1) * FOUT];
        c = __builtin_amdgcn_wmma_f32_16x16x4_f32(
                /*neg_a=*/false, a, /*neg_b=*/false, b,
                /*c_mod=*/(short)0, c, /*reuse_a=*/false, /*reuse_b=*/false);
    }

    #pragma unroll
    for (int i = 0; i < 8; ++i) {
        int g = row0 + rt * 16 + i + half * 8;
        if (g < n) out[(size_t)g * FOUT + ct * 16 + l15] = c[i];
    }
}

// ---------------- acc = bias + self-loop term ----------------

template<int F>
__global__ void init_acc_kernel(float* __restrict__ acc, const float* __restrict__ hw,
                                const float* __restrict__ bias,
                                const float* __restrict__ dinv, int n) {
    int i = blockIdx.x * blockDim.x + threadIdx.x;
    if (i >= n * F) return;
    int node = i / F;
    int f = i - node * F;
    float di = dinv[node];
    acc[i] = bias[f] + hw[i] * di * di;     // self-loop: norm = dinv[i]*dinv[i]
}

// ---------------- edge gather/scale/scatter-add ----------------
// One wave per edge; lane l handles F/32 consecutive features (b128 coalesced gather,
// per-float global_atomic_add_f32 into L2-resident accumulator).

template<int F>
__global__ void edge_scatter_kernel(const float* __restrict__ hw, const int* __restrict__ src,
                                    const int* __restrict__ dst, const float* __restrict__ dinv,
                                    float* __restrict__ acc, int E) {
    constexpr int VEC = F / 32;
    int t = blockIdx.x * blockDim.x + threadIdx.x;
    int e = t >> 5;
    if (e >= E) return;
    int lane = t & 31;
    int s = src[e], d = dst[e];
    float nw = dinv[s] * dinv[d];
    const float* hp = hw  + (size_t)s * F + lane * VEC;
    float*       ap = acc + (size_t)d * F + lane * VEC;
    float v[VEC];
    #pragma unroll
    for (int j = 0; j < VEC; ++j) v[j] = hp[j];
    #pragma unroll
    for (int j = 0; j < VEC; ++j) atomicAdd(ap + j, v[j] * nw);
}

// ---------------- ReLU into next layer's activation buffer ----------------

__global__ void relu_kernel(const float* __restrict__ in, float* __restrict__ out, int count) {
    int i = blockIdx.x * blockDim.x + threadIdx.x;
    if (i < count) out[i] = fmaxf(in[i], 0.0f);
}

// ---------------- launch ----------------

extern "C" void kernel_launch(void* const* d_in, const int* in_sizes, int n_in,
                              void* d_out, int out_size, void* d_ws, size_t ws_size,
                              hipStream_t stream) {
    (void)n_in; (void)out_size; (void)ws_size;
    const float* x  = (const float*)d_in[0];
    const float* W1 = (const float*)d_in[1];
    const float* b1 = (const float*)d_in[2];
    const float* W2 = (const float*)d_in[3];
    const float* b2 = (const float*)d_in[4];
    const float* W3 = (const float*)d_in[5];
    const float* b3 = (const float*)d_in[6];
    const int*   ei = (const int*)d_in[7];

    const int n = in_sizes[0] / D_IN;       // 50000
    const int E = in_sizes[7] / 2;          // 1.6M
    const int* src = ei;
    const int* dst = ei + E;

    float* ws   = (float*)d_ws;
    float* dinv = ws;                                  // [n]
    float* bufA = ws + n;                              // [n,128] activations
    float* bufB = bufA + (size_t)n * 128;              // [n,128] GEMM output hW
    float* bufC = bufB + (size_t)n * 128;              // [n,128] aggregation acc
    float* out  = (float*)d_out;                       // [n,64]

    const int TB = 256;
    const int nBlkN    = (n + TB - 1) / TB;
    const int nBlkE    = (E + TB - 1) / TB;
    const int nBlkNF   = (n * 128 + TB - 1) / TB;
    const int nBlkNF64 = (n * 64  + TB - 1) / TB;
    const int nBlkEdge = (int)(((size_t)E * 32 + TB - 1) / TB);

    // normalization: deg = indeg + 1, dinv = rsqrt(deg)
    deg_init_kernel<<<nBlkN, TB, 0, stream>>>(dinv, n);
    deg_scatter_kernel<<<nBlkE, TB, 0, stream>>>(dst, dinv, E);
    rsqrt_kernel<<<nBlkN, TB, 0, stream>>>(dinv, n);

    // ---- layer 1: x -> bufA (ReLU) ----
    gcn_gemm_wmma<128,128><<<(n + 15) / 16, 256, 0, stream>>>(x, W1, bufB, n);
    init_acc_kernel<128><<<nBlkNF, TB, 0, stream>>>(bufC, bufB, b1, dinv, n);
    edge_scatter_kernel<128><<<nBlkEdge, TB, 0, stream>>>(bufB, src, dst, dinv, bufC, E);
    relu_kernel<<<nBlkNF, TB, 0, stream>>>(bufC, bufA, n * 128);

    // ---- layer 2: bufA -> bufA (ReLU) ----
    gcn_gemm_wmma<128,128><<<(n + 15) / 16, 256, 0, stream>>>(bufA, W2, bufB, n);
    init_acc_kernel<128><<<nBlkNF, TB, 0, stream>>>(bufC, bufB, b2, dinv, n);
    edge_scatter_kernel<128><<<nBlkEdge, TB, 0, stream>>>(bufB, src, dst, dinv, bufC, E);
    relu_kernel<<<nBlkNF, TB, 0, stream>>>(bufC, bufA, n * 128);

    // ---- layer 3: bufA -> d_out (no ReLU) ----
    gcn_gemm_wmma<128,64><<<(n + 31) / 32, 256, 0, stream>>>(bufA, W3, bufB, n);
    init_acc_kernel<64><<<nBlkNF64, TB, 0, stream>>>(out, bufB, b3, dinv, n);
    edge_scatter_kernel<64><<<nBlkEdge, TB, 0, stream>>>(bufB, src, dst, dinv, out, E);
}